// WhdrTestLoss_Paper_15994458211238
// MI455X (gfx1250) — compile-verified
//
#include <hip/hip_runtime.h>
#include <hip/hip_bf16.h>

#define TPB 256
#define BPI 64   // blocks per image

typedef __attribute__((ext_vector_type(2))) float v2f;
typedef __attribute__((ext_vector_type(8))) float v8f;

// Exact f32 wave32 sum using V_WMMA_F32_16X16X4_F32.
// A (16x4 f32) layout per ISA: vgpr0 = {lanes0-15: K=0, lanes16-31: K=2},
//                              vgpr1 = {lanes0-15: K=1, lanes16-31: K=3}.
// With a = {x,x}:  A[m][0]=A[m][1]=x_m (m = lane<16), A[m][2]=A[m][3]=x_{m+16}.
// With B = all 0.5: D[m][n] = 0.5*(2*x_m + 2*x_{m+16}) = x_m + x_{m+16}.
// C/D layout: vgpr j at lane L holds D[j + (L<16?0:8)][L%16].
// Sum of a lane's 8 D vgprs = sum over 8 consecutive m of (x_m + x_{m+16});
// adding the cross-half partner (shfl_xor 16) yields the full 32-lane sum.
__device__ __forceinline__ float wave_sum_wmma(float x) {
    v2f a; a[0] = x;    a[1] = x;
    v2f b; b[0] = 0.5f; b[1] = 0.5f;
    v8f c = {};
    v8f d = __builtin_amdgcn_wmma_f32_16x16x4_f32(
        /*neg_a=*/false, a, /*neg_b=*/false, b,
        /*c_mod=*/(short)0, c, /*reuse_a=*/false, /*reuse_b=*/false);
    float s = ((d[0] + d[1]) + (d[2] + d[3])) + ((d[4] + d[5]) + (d[6] + d[7]));
    s += __shfl_xor(s, 16, 32);
    return s;   // total of x across the wave, in every lane
}

__global__ __launch_bounds__(TPB) void whdr_partial_kernel(
    const float* __restrict__ img,      // B * H * W
    const float* __restrict__ cmp,      // B * M * 6
    const int*  __restrict__ numComp,   // B
    float2*     __restrict__ partial,   // B * gridDim.x
    int M, int W)
{
    const int b = blockIdx.y;
    const float* __restrict__ im = img + (size_t)b * (size_t)W * (size_t)W;
    const float* __restrict__ cb = cmp + (size_t)b * (size_t)M * 6u;
    const int n = numComp[b];                 // tail has weight 0 in reference
    const int stride = gridDim.x * TPB;

    const float EPS = 1e-10f;
    const float THR = 1.0f + 0.1f;

    float num = 0.0f, den = 0.0f;

    for (int i = blockIdx.x * TPB + threadIdx.x; i < n; i += stride) {
        const float* pc = cb + (size_t)i * 6u;
        // prefetch next grid-stride element of the stream (speculative OK)
        __builtin_prefetch(cb + (size_t)(i + stride) * 6u, 0, 0);

        // 24B record: three aligned 8B non-temporal loads (don't pollute L2,
        // which we want to keep holding the gathered images)
        v2f c0 = __builtin_nontemporal_load((const v2f*)pc);        // x1,y1
        v2f c1 = __builtin_nontemporal_load((const v2f*)(pc + 2));  // x2,y2
        v2f c2 = __builtin_nontemporal_load((const v2f*)(pc + 4));  // darker,weight

        int x1 = (int)c0[0], y1 = (int)c0[1];
        int x2 = (int)c1[0], y2 = (int)c1[1];
        int darker = (int)c2[0];
        float w = c2[1];

        float r1 = im[y1 * W + x1];
        float r2 = im[y2 * W + x2];

        // match reference arithmetic exactly (divisions, not mul-compare)
        int alg = (r2 / (r1 + EPS) > THR) ? 1
                : ((r1 / (r2 + EPS) > THR) ? 2 : 0);

        float mism = (alg != darker) ? 1.0f : 0.0f;
        num = fmaf(w, mism, num);
        den += w;
    }

    // wave32 reduction via FP32 WMMA (EXEC is full here: uniform flow, full blocks)
    float wnum = wave_sum_wmma(num);
    float wden = wave_sum_wmma(den);

    __shared__ float2 wsum[TPB / 32];
    const int lane = threadIdx.x & 31;
    const int wid  = threadIdx.x >> 5;
    if (lane == 0) wsum[wid] = make_float2(wnum, wden);
    __syncthreads();

    if (threadIdx.x == 0) {
        float2 acc = wsum[0];
        #pragma unroll
        for (int k = 1; k < TPB / 32; ++k) { acc.x += wsum[k].x; acc.y += wsum[k].y; }
        partial[(size_t)b * gridDim.x + blockIdx.x] = acc;
    }
}

// One block, 32 threads (one wave) per image: fixed-order, deterministic.
__global__ void whdr_final_kernel(const float2* __restrict__ partial,
                                  float* __restrict__ out, int bpi, int B)
{
    const int t = threadIdx.x;
    const int b = t >> 5;       // image = wave index (threads 32b..32b+31)
    const int lane = t & 31;

    float num = 0.0f, den = 0.0f;
    if (b < B) {
        for (int j = lane; j < bpi; j += 32) {
            float2 p = partial[(size_t)b * bpi + j];
            num += p.x; den += p.y;
        }
    }
    #pragma unroll
    for (int off = 16; off > 0; off >>= 1) {
        num += __shfl_xor(num, off, 32);
        den += __shfl_xor(den, off, 32);
    }

    __shared__ float ratios[64];
    if (b < B && lane == 0) ratios[b] = num / den;
    __syncthreads();

    if (t == 0) {
        float s = 0.0f;
        for (int k = 0; k < B; ++k) s += ratios[k];
        out[0] = s / (float)B;
    }
}

extern "C" void kernel_launch(void* const* d_in, const int* in_sizes, int n_in,
                              void* d_out, int out_size, void* d_ws, size_t ws_size,
                              hipStream_t stream) {
    const float* img  = (const float*)d_in[0];   // (B,1,H,W) f32
    const float* cmp  = (const float*)d_in[1];   // (B,M,6)   f32
    const int*   ncmp = (const int*)  d_in[2];   // (B,)      i32
    float*       out  = (float*)d_out;

    const int B = in_sizes[2];
    const long HW = (long)in_sizes[0] / B;
    int W = 1;
    while ((long)W * W < HW) ++W;                // square image (512)
    const int M = in_sizes[1] / (B * 6);

    float2* partial = (float2*)d_ws;             // B * BPI float2 (8 KB) fully
                                                 // overwritten before read

    dim3 grid(BPI, B);
    whdr_partial_kernel<<<grid, TPB, 0, stream>>>(img, cmp, ncmp, partial, M, W);
    whdr_final_kernel<<<1, B * 32, 0, stream>>>(partial, out, BPI, B);
}